// HalfEdgeConv_23802708755009
// MI455X (gfx1250) — compile-verified
//
#include <hip/hip_runtime.h>
#include <hip/hip_bf16.h>

typedef __attribute__((ext_vector_type(2))) float v2f;
typedef __attribute__((ext_vector_type(8))) float v8f;

#define CIN      64
#define KNB      5
#define COUT     64
#define KSTEPS   80              // 320 / 4 : WMMA K-steps (K=4 each)
#define NTILES   4               // 64 / 16 : N-tiles of 16 cols
#define MTILES   2               // M-tiles (16 edges each) per wave
#define WAVE_EDGES (16 * MTILES) // 32 edges per wave
#define BLK_EDGES  (4 * WAVE_EDGES) // 128 edges per block (4 waves)
#define LDSW     (KSTEPS * NTILES * 32 * 2)  // 20480 floats = 80 KB

// Each wave computes a 32(M) x 64(N) tile of out = ReLU(gather(x)@W + b)
// with V_WMMA_F32_16X16X4_F32. Two M-tiles share every B fragment, halving
// LDS traffic per WMMA. A fragments are gathered straight from global
// (one b64/lane per K-step per M-tile); W is staged once per block into LDS,
// pre-swizzled so each B fragment is a single lane-contiguous ds_load_b64.
__global__ __launch_bounds__(128)
void HalfEdgeConv_wmma_kernel(const float* __restrict__ x,
                              const int*   __restrict__ nbr,
                              const float* __restrict__ W,
                              const float* __restrict__ bias,
                              float*       __restrict__ out,
                              int E, int numTiles)
{
    __shared__ float lds_w[LDSW];

    const int tid = threadIdx.x;

    // ---- Stage W into LDS, swizzled so B fragment (s, n) is one b64/lane ----
    // lds index i = s*256 + n*64 + lane*2 + j
    // lane<16 : pair = { W[4s+0][n*16+lane],    W[4s+1][n*16+lane]    }
    // lane>=16: pair = { W[4s+2][n*16+lane-16], W[4s+3][n*16+lane-16] }
    for (int i = tid; i < LDSW; i += 128) {
        const int j    = i & 1;
        const int lane = (i >> 1) & 31;
        const int n    = (i >> 6) & 3;
        const int s    = i >> 8;
        const int row  = 4 * s + 2 * (lane >> 4) + j;
        const int col  = n * 16 + (lane & 15);
        lds_w[i] = W[row * COUT + col];
    }
    __syncthreads();

    const int wave = tid >> 5;
    const int lane = tid & 31;
    const int half = lane >> 4;   // 0: lanes 0-15, 1: lanes 16-31
    const int l15  = lane & 15;

    // Bias per N-tile: C/D element's column is n*16 + (lane&15) in every VGPR.
    float bv[NTILES];
#pragma unroll
    for (int n = 0; n < NTILES; ++n) bv[n] = bias[n * 16 + l15];

    for (int t = blockIdx.x; t < numTiles; t += gridDim.x) {
        const int e0 = t * BLK_EDGES + wave * WAVE_EDGES;

        // Gathered A-row base pointers for both M-tiles (fold in the +2 column
        // offset for the upper half-wave, which supplies K=k0+2 / k0+3).
        const float* base[MTILES][KNB];
#pragma unroll
        for (int m = 0; m < MTILES; ++m) {
            const int row = e0 + m * 16 + l15;
            const int rc  = row < E ? row : 0;   // clamp: EXEC must stay full
#pragma unroll
            for (int j = 0; j < KNB; ++j) {
                const long idx = (long)nbr[rc * KNB + j];
                base[m][j] = x + idx * CIN + half * 2;
            }
        }

        // Accumulators initialized to bias (free bias add).
        v8f acc[MTILES][NTILES];
#pragma unroll
        for (int m = 0; m < MTILES; ++m)
#pragma unroll
            for (int n = 0; n < NTILES; ++n)
#pragma unroll
                for (int c = 0; c < 8; ++c) acc[m][n][c] = bv[n];

        // Main K loop: 5 neighbor segments x 16 sub-steps of K=4.
        // Per sub-step: 2 gathered global b64 + 4 ds b64 + 8 WMMA.
#pragma unroll
        for (int j = 0; j < KNB; ++j) {
#pragma unroll 4
            for (int ss = 0; ss < 16; ++ss) {
                const int s = j * 16 + ss;
                v2f a[MTILES];
#pragma unroll
                for (int m = 0; m < MTILES; ++m)
                    a[m] = *(const v2f*)(base[m][j] + ss * 4);
#pragma unroll
                for (int n = 0; n < NTILES; ++n) {
                    const v2f bf = *(const v2f*)&lds_w[(s * 4 + n) * 64 + lane * 2];
#pragma unroll
                    for (int m = 0; m < MTILES; ++m) {
                        acc[m][n] = __builtin_amdgcn_wmma_f32_16x16x4_f32(
                            /*neg_a=*/false, a[m], /*neg_b=*/false, bf,
                            /*c_mod=*/(short)0, acc[m][n],
                            /*reuse_a=*/false, /*reuse_b=*/false);
                    }
                }
            }
        }

        // Epilogue: ReLU + store. C/D layout: VGPR v, lanes 0-15 -> M=v,
        // lanes 16-31 -> M=v+8; column = n*16 + (lane&15).
#pragma unroll
        for (int m = 0; m < MTILES; ++m) {
#pragma unroll
            for (int n = 0; n < NTILES; ++n) {
#pragma unroll
                for (int v = 0; v < 8; ++v) {
                    const int r = e0 + m * 16 + v + 8 * half;
                    if (r < E)
                        out[(long)r * COUT + n * 16 + l15] = fmaxf(acc[m][n][v], 0.0f);
                }
            }
        }
    }
}

extern "C" void kernel_launch(void* const* d_in, const int* in_sizes, int n_in,
                              void* d_out, int out_size, void* d_ws, size_t ws_size,
                              hipStream_t stream) {
    const float* x    = (const float*)d_in[0];
    const int*   nbr  = (const int*)  d_in[1];
    const float* W    = (const float*)d_in[2];
    const float* bias = (const float*)d_in[3];
    float*       out  = (float*)d_out;

    const int E        = in_sizes[0] / CIN;            // 1,000,000
    const int numTiles = (E + BLK_EDGES - 1) / BLK_EDGES;  // 7813

    int blocks = numTiles < 2048 ? numTiles : 2048;    // grid-stride amortizes W staging
    hipLaunchKernelGGL(HalfEdgeConv_wmma_kernel, dim3(blocks), dim3(128), 0, stream,
                       x, nbr, W, bias, out, E, numTiles);
}